// Perspective_Layer_3350074491126
// MI455X (gfx1250) — compile-verified
//
#include <hip/hip_runtime.h>

// CDNA5 / gfx1250. Wave32. WMMA for the projective-transform GEMM stage.

typedef float v2f __attribute__((ext_vector_type(2)));
typedef float v8f __attribute__((ext_vector_type(8)));

#define TMM   2
#define CCH   48
#define NBATCH 8
#define HH    224
#define WW    224
#define PP    (HH * WW)      // 50176 pixels
#define NTC   (TMM * CCH)    // 96 output channels
#define NROWS (NTC * 3)      // 288 rows of theta_ext (t,c,i)
#define NPIX  32             // pixels handled per workgroup
#define LPITCH 33            // LDS row pitch (dwords) -> bank-conflict-free

__global__ __launch_bounds__(192)
void persp_warp_kernel(const float* __restrict__ x,     // (B,H,W,C)
                       const float* __restrict__ wt,    // (TM,C,8)
                       float* __restrict__ out)         // (B,Ho,Wo,TM*C)
{
    __shared__ float bg[NROWS * LPITCH];   // 288*33*4 = 38016 B

    const int tid  = threadIdx.x;
    const int lane = tid & 31;
    const int wave = tid >> 5;             // 0..5 (6 waves of 32)
    const int p0   = blockIdx.x * NPIX;    // first pixel of this block

    // ================= Phase 1 =================
    // bg(288 x 32) = A(288 x 4) @ G(4 x 32), K=3 padded to 4 with zeros.
    // A row r = (t*C+c)*3 + i ; A[r][j] = theta[t][c][i][j] (j<3), 0 (j==3)
    // theta flat k=i*3+j: k<8 -> wt[tc*8+k], k==8 -> 1.0 (appended ones).
    // 18 M-tiles of 16 rows x 2 N-tiles of 16 pixels; wave w owns M-tiles 3w..3w+2.
    //
    // V_WMMA operand layout (f32, wave32):
    //   A 16x4 : lanes 0-15 -> {K=0,K=1}, lanes 16-31 -> {K=2,K=3}, M = lane&15
    //   B 4x16 : lanes 0-15 -> {K=0,K=1}, lanes 16-31 -> {K=2,K=3}, N = lane&15
    //   D 16x16: VGPR i -> row M=i (lanes 0-15) / M=i+8 (lanes 16-31), N = lane&15
    const int half = lane >> 4;            // 0 or 1
    const int lr   = lane & 15;

    for (int mtl = 0; mtl < 3; ++mtl) {
        const int mt  = wave * 3 + mtl;    // 0..17
        const int row = mt * 16 + lr;      // < 288
        const int tcr = row / 3;
        const int ir  = row % 3;
        // Branchless A-operand fetch (keeps EXEC all-ones, no divergent loads):
        //   half==0: k0 = 3i   in {0,3,6}, k1 = 3i+1 in {1,4,7}  -> plain loads
        //   half==1: k0 = 3i+2 in {2,5,8} -> clamp load idx, select 1.0 at k0==8
        //            k1 = pad  -> 0.0
        const int k0  = ir * 3 + (half ? 2 : 0);
        const int k1  = ir * 3 + 1;                     // always < 8
        const float w0 = wt[tcr * 8 + min(k0, 7)];      // unconditional load
        const float w1 = wt[tcr * 8 + k1];              // unconditional load
        const float a0 = (k0 == 8) ? 1.0f : w0;         // v_cndmask, no branch
        const float a1 = half ? 0.0f : w1;              // v_cndmask, no branch
        v2f av; av.x = a0; av.y = a1;

        #pragma unroll
        for (int nt = 0; nt < 2; ++nt) {
            const int p  = p0 + nt * 16 + lr;
            const float gx = -1.0f + (2.0f / 223.0f) * (float)(p % WW);
            const float gy = -1.0f + (2.0f / 223.0f) * (float)(p / WW);
            // B rows: K0 = xg, K1 = yg, K2 = 1, K3 = 0 (pad)
            v2f bv;
            bv.x = half ? 1.0f : gx;
            bv.y = half ? 0.0f : gy;
            v8f d = {};
            d = __builtin_amdgcn_wmma_f32_16x16x4_f32(
                    /*neg_a=*/false, av, /*neg_b=*/false, bv,
                    /*c_mod=*/(short)0, d, /*reuse_a=*/false, /*reuse_b=*/false);
            const int rbase = mt * 16 + (half ? 8 : 0);
            const int pc    = nt * 16 + lr;
            #pragma unroll
            for (int r = 0; r < 8; ++r)
                bg[(rbase + r) * LPITCH + pc] = d[r];
        }
    }
    __syncthreads();

    // ================= Phase 2 =================
    // 96 channels x 32 pixels per block. Lanes are channel-contiguous so each
    // wave's stores form contiguous 128B bursts of out[(b*P+p)*96 + tc].
    const int tc   = tid % NTC;            // 0..95, contiguous within a wave
    const int psub = tid / NTC;            // 0 or 1
    const int ch   = tc % CCH;             // input channel

    for (int it = 0; it < NPIX / 2; ++it) {
        const int pl = psub * 16 + it;     // local pixel, uniform per wave
        const int p  = p0 + pl;

        const float bx = bg[(tc * 3 + 0) * LPITCH + pl];
        const float by = bg[(tc * 3 + 1) * LPITCH + pl];
        const float om = bg[(tc * 3 + 2) * LPITCH + pl];
        const float inv = 1.0f / om;
        const float xf = 0.5f * (bx * inv + 1.0f) * 222.0f;  // (W-2)
        const float yf = 0.5f * (by * inv + 1.0f) * 222.0f;  // (H-2)

        const int x0 = (int)floorf(xf);
        const int y0 = (int)floorf(yf);
        const int x0c = min(max(x0,     0), WW - 1);
        const int x1c = min(max(x0 + 1, 0), WW - 1);
        const int y0c = min(max(y0,     0), HH - 1);
        const int y1c = min(max(y0 + 1, 0), HH - 1);

        const float x0f = (float)x0c, x1f = (float)x1c;
        const float y0f = (float)y0c, y1f = (float)y1c;
        const float wa = (x1f - xf) * (y1f - yf);
        const float wb = (x1f - xf) * (yf - y0f);
        const float wc = (xf - x0f) * (y1f - yf);
        const float wd = (xf - x0f) * (yf - y0f);

        const int i00 = (y0c * WW + x0c) * CCH + ch;   // Ia
        const int i10 = (y1c * WW + x0c) * CCH + ch;   // Ib
        const int i01 = (y0c * WW + x1c) * CCH + ch;   // Ic
        const int i11 = (y1c * WW + x1c) * CCH + ch;   // Id

        #pragma unroll
        for (int b = 0; b < NBATCH; ++b) {
            const float* xb = x + (size_t)b * (PP * CCH);   // L2-resident (77MB < 192MB)
            const float v = wa * xb[i00] + wb * xb[i10]
                          + wc * xb[i01] + wd * xb[i11];
            out[((size_t)b * PP + p) * NTC + tc] = v;       // coalesced per wave
        }
    }
}

extern "C" void kernel_launch(void* const* d_in, const int* in_sizes, int n_in,
                              void* d_out, int out_size, void* d_ws, size_t ws_size,
                              hipStream_t stream) {
    (void)in_sizes; (void)n_in; (void)out_size; (void)d_ws; (void)ws_size;
    const float* x  = (const float*)d_in[0];   // (8,224,224,48) f32
    const float* wt = (const float*)d_in[1];   // (2,48,8) f32
    float* out = (float*)d_out;                // (8,224,224,96) f32

    dim3 grid(PP / NPIX);                      // 1568 blocks
    dim3 block(192);                           // 6 wave32 waves
    persp_warp_kernel<<<grid, block, 0, stream>>>(x, wt, out);
}